// GATLSTM_19224273617373
// MI455X (gfx1250) — compile-verified
//
#include <hip/hip_runtime.h>
#include <math.h>

#define NN 50000
#define NE 1600000
#define E2 (NE + NN)          // edges + self loops
#define HC 128                // HEADS*GAT_OUT
#define NG 64                 // NUM_GRAPHS
#define NCLS 4

typedef float v2f __attribute__((ext_vector_type(2)));
typedef float v8f __attribute__((ext_vector_type(8)));

// ---------------- f32 WMMA GEMM: out[NN,128] = X[NN,128] @ W[128,128] + b ----
__global__ __launch_bounds__(256) void gemm_xw(const float* __restrict__ X,
                                               const float* __restrict__ W,
                                               const float* __restrict__ b,
                                               float* __restrict__ out) {
  const int lane = threadIdx.x & 31;
  const int wave = threadIdx.x >> 5;
  const int rowBase = blockIdx.x * 16;
  const int n0 = wave * 16;               // 8 waves cover 128 output cols
  const int mrow = rowBase + (lane & 15); // A: lanes 0-15 M, 16-31 same M, K+2
  const int col  = n0 + (lane & 15);      // B/C/D: N striped over lanes
  const int khi  = (lane >> 4) * 2;       // 0 or 2

  v8f acc = {};
  #pragma unroll 8
  for (int k0 = 0; k0 < 128; k0 += 4) {
    const int kk = k0 + khi;
    const float2 av = *(const float2*)(X + (size_t)mrow * 128 + kk);
    v2f a; a.x = av.x; a.y = av.y;
    v2f bb;
    bb.x = W[(size_t)kk * 128 + col];
    bb.y = W[(size_t)(kk + 1) * 128 + col];
    acc = __builtin_amdgcn_wmma_f32_16x16x4_f32(false, a, false, bb,
                                                (short)0, acc, false, false);
  }
  // C/D layout: VGPR i: lanes 0-15 -> M=i, lanes 16-31 -> M=8+i
  const int rbase = rowBase + ((lane >> 4) << 3);
  const float bias = b[col];
  #pragma unroll
  for (int i = 0; i < 8; i++)
    out[(size_t)(rbase + i) * 128 + col] = acc[i] + bias;
}

// ---------------- init workspace -------------------------------------------
__global__ void init_ws(float* __restrict__ accum, float* __restrict__ m,
                        float* __restrict__ denom, float* __restrict__ gsum,
                        float* __restrict__ gcnt) {
  const int i = blockIdx.x * blockDim.x + threadIdx.x;
  accum[i] = 0.f;                       // grid sized exactly NN*128
  if (i < NN * 2) { m[i] = -3.0e38f; denom[i] = 0.f; }
  if (i < NG * 64) gsum[i] = 0.f;
  if (i < NG) gcnt[i] = 0.f;
}

__device__ __forceinline__ void atomicMaxF(float* addr, float val) {
  if (val >= 0.f) atomicMax((int*)addr, __float_as_int(val));
  else            atomicMin((unsigned int*)addr, (unsigned int)__float_as_int(val));
}

__device__ __forceinline__ void get_edge(const int* __restrict__ ei, int e,
                                         int& src, int& dst) {
  if (e < NE) { src = ei[e]; dst = ei[NE + e]; }
  else        { src = dst = e - NE; }   // self loop
}

// ---------------- pass 1: e = att . lrelu(xl[src]+xr[dst]); segment max -----
__global__ void edge_attn(const int* __restrict__ ei, const float* __restrict__ xl,
                          const float* __restrict__ xr, const float* __restrict__ att,
                          float* __restrict__ ebuf, float* __restrict__ m) {
  const long long idx = (long long)blockIdx.x * blockDim.x + threadIdx.x;
  if (idx >= (long long)E2 * 2) return;
  const int e = (int)(idx >> 1), h = (int)(idx & 1);
  int src, dst; get_edge(ei, e, src, dst);
  const float4* xls = (const float4*)(xl + (size_t)src * 128 + h * 64);
  const float4* xrd = (const float4*)(xr + (size_t)dst * 128 + h * 64);
  const float4* at4 = (const float4*)(att + h * 64);
  float acc = 0.f;
  #pragma unroll
  for (int i = 0; i < 16; i++) {
    const float4 l = xls[i], r = xrd[i], a = at4[i];
    float s;
    s = l.x + r.x; acc += (s > 0.f ? s : 0.2f * s) * a.x;
    s = l.y + r.y; acc += (s > 0.f ? s : 0.2f * s) * a.y;
    s = l.z + r.z; acc += (s > 0.f ? s : 0.2f * s) * a.z;
    s = l.w + r.w; acc += (s > 0.f ? s : 0.2f * s) * a.w;
  }
  ebuf[idx] = acc;
  atomicMaxF(&m[dst * 2 + h], acc);
}

// ---------------- pass 2: a = exp(e - m[dst]); segment sum ------------------
__global__ void edge_soft(const int* __restrict__ ei, float* __restrict__ ebuf,
                          const float* __restrict__ m, float* __restrict__ denom) {
  const long long idx = (long long)blockIdx.x * blockDim.x + threadIdx.x;
  if (idx >= (long long)E2 * 2) return;
  const int e = (int)(idx >> 1), h = (int)(idx & 1);
  int src, dst; get_edge(ei, e, src, dst);
  const float a = expf(ebuf[idx] - m[dst * 2 + h]);
  ebuf[idx] = a;                        // reuse buffer in place
  atomicAdd(&denom[dst * 2 + h], a);
}

// ---------------- pass 3: accum[dst] += alpha * xl[src] ---------------------
__global__ void edge_scat(const int* __restrict__ ei, const float* __restrict__ ebuf,
                          const float* __restrict__ denom, const float* __restrict__ xl,
                          float* __restrict__ accum) {
  const long long idx = (long long)blockIdx.x * blockDim.x + threadIdx.x;
  if (idx >= (long long)E2 * 2) return;
  const int e = (int)(idx >> 1), h = (int)(idx & 1);
  int src, dst; get_edge(ei, e, src, dst);
  const float alpha = ebuf[idx] / denom[dst * 2 + h];
  const float4* xls = (const float4*)(xl + (size_t)src * 128 + h * 64);
  float* ac = accum + (size_t)dst * 128 + h * 64;
  #pragma unroll
  for (int i = 0; i < 16; i++) {
    const float4 l = xls[i];
    atomicAdd(ac + 4 * i + 0, alpha * l.x);
    atomicAdd(ac + 4 * i + 1, alpha * l.y);
    atomicAdd(ac + 4 * i + 2, alpha * l.z);
    atomicAdd(ac + 4 * i + 3, alpha * l.w);
  }
}

// ---------------- head mean + bias, then global mean pool -------------------
__global__ void pool_k(const float* __restrict__ accum, const float* __restrict__ gbias,
                       const int* __restrict__ batch, float* __restrict__ gsum,
                       float* __restrict__ gcnt) {
  const int i = blockIdx.x * blockDim.x + threadIdx.x;   // NN*64 exact
  const int node = i >> 6, c = i & 63;
  const float v = 0.5f * (accum[(size_t)node * 128 + c] +
                          accum[(size_t)node * 128 + 64 + c]) + gbias[c];
  const int b = batch[node];
  atomicAdd(&gsum[b * 64 + c], v);
  if (c == 0) atomicAdd(&gcnt[b], 1.0f);
}

// ---------------- BiLSTM (len-1 seq, h0=c0=0) + FC --------------------------
__device__ __forceinline__ float sigm(float x) { return 1.f / (1.f + expf(-x)); }

__global__ __launch_bounds__(256) void head_k(
    const float* __restrict__ gsum, const float* __restrict__ gcnt,
    const float* __restrict__ Wih0, const float* __restrict__ bih0,
    const float* __restrict__ bhh0, const float* __restrict__ Wih1,
    const float* __restrict__ bih1, const float* __restrict__ bhh1,
    const float* __restrict__ fcw,  const float* __restrict__ fcb,
    float* __restrict__ out) {
  __shared__ float semb[64];
  __shared__ float sg[1024];
  __shared__ float sx1[256];
  __shared__ float sh1[256];
  const int b = blockIdx.x, t = threadIdx.x;

  if (t < 64) semb[t] = gsum[b * 64 + t] / fmaxf(gcnt[b], 1.0f);
  __syncthreads();

  // layer0 gates: 2 dirs x 512 gates, K=64
  #pragma unroll
  for (int q = 0; q < 4; q++) {
    const int gi = t + q * 256, d = gi >> 9, j = gi & 511;
    float acc = bih0[d * 512 + j] + bhh0[d * 512 + j];
    const float* w = Wih0 + ((size_t)d * 512 + j) * 64;
    for (int k = 0; k < 64; k++) acc += semb[k] * w[k];
    sg[gi] = acc;
  }
  __syncthreads();
  {
    const int d = t >> 7, j = t & 127, base = d * 512;
    const float ig = sg[base + j], gg = sg[base + 256 + j], og = sg[base + 384 + j];
    const float c = sigm(ig) * tanhf(gg);
    sx1[t] = sigm(og) * tanhf(c);       // x1 = [hf0 | hb0]
  }
  __syncthreads();

  // layer1 gates: K=256
  #pragma unroll
  for (int q = 0; q < 4; q++) {
    const int gi = t + q * 256, d = gi >> 9, j = gi & 511;
    float acc = bih1[d * 512 + j] + bhh1[d * 512 + j];
    const float* w = Wih1 + ((size_t)d * 512 + j) * 256;
    for (int k = 0; k < 256; k++) acc += sx1[k] * w[k];
    sg[gi] = acc;
  }
  __syncthreads();
  {
    const int d = t >> 7, j = t & 127, base = d * 512;
    const float ig = sg[base + j], gg = sg[base + 256 + j], og = sg[base + 384 + j];
    const float c = sigm(ig) * tanhf(gg);
    sh1[t] = sigm(og) * tanhf(c);       // hcat = [hf1 | hb1]
  }
  __syncthreads();

  if (t < NCLS) {
    float acc = fcb[t];
    const float* w = fcw + t * 256;
    for (int k = 0; k < 256; k++) acc += sh1[k] * w[k];
    out[b * NCLS + t] = acc;
  }
}

extern "C" void kernel_launch(void* const* d_in, const int* in_sizes, int n_in,
                              void* d_out, int out_size, void* d_ws, size_t ws_size,
                              hipStream_t stream) {
  const float* x    = (const float*)d_in[0];
  const int*   ei   = (const int*)d_in[1];
  const int*   bidx = (const int*)d_in[2];
  const float* Wl   = (const float*)d_in[3];
  const float* bl   = (const float*)d_in[4];
  const float* Wr   = (const float*)d_in[5];
  const float* br   = (const float*)d_in[6];
  const float* att  = (const float*)d_in[7];
  const float* gb   = (const float*)d_in[8];
  const float* Wih0 = (const float*)d_in[9];
  const float* bih0 = (const float*)d_in[11];
  const float* bhh0 = (const float*)d_in[12];
  const float* Wih1 = (const float*)d_in[13];
  const float* bih1 = (const float*)d_in[15];
  const float* bhh1 = (const float*)d_in[16];
  const float* fcw  = (const float*)d_in[17];
  const float* fcb  = (const float*)d_in[18];
  float* out = (float*)d_out;

  float* ws    = (float*)d_ws;
  float* xl    = ws;
  float* xr    = xl + (size_t)NN * 128;
  float* ebuf  = xr + (size_t)NN * 128;
  float* m     = ebuf + (size_t)E2 * 2;
  float* denom = m + (size_t)NN * 2;
  float* accum = denom + (size_t)NN * 2;
  float* gsum  = accum + (size_t)NN * 128;
  float* gcnt  = gsum + NG * 64;

  gemm_xw<<<NN / 16, 256, 0, stream>>>(x, Wl, bl, xl);
  gemm_xw<<<NN / 16, 256, 0, stream>>>(x, Wr, br, xr);
  init_ws<<<(NN * 128) / 256, 256, 0, stream>>>(accum, m, denom, gsum, gcnt);

  const int tE = E2 * 2;
  const int gE = (tE + 255) / 256;
  edge_attn<<<gE, 256, 0, stream>>>(ei, xl, xr, att, ebuf, m);
  edge_soft<<<gE, 256, 0, stream>>>(ei, ebuf, m, denom);
  edge_scat<<<gE, 256, 0, stream>>>(ei, ebuf, denom, xl, accum);

  pool_k<<<(NN * 64) / 256, 256, 0, stream>>>(accum, gb, bidx, gsum, gcnt);
  head_k<<<NG, 256, 0, stream>>>(gsum, gcnt, Wih0, bih0, bhh0,
                                 Wih1, bih1, bhh1, fcw, fcb, out);
}